// SafeDrugModel_86234353369548
// MI455X (gfx1250) — compile-verified
//
#include <hip/hip_runtime.h>
#include <math.h>

typedef __attribute__((ext_vector_type(2))) float v2f;
typedef __attribute__((ext_vector_type(8))) float v8f;

// ---------- helpers ----------
__device__ __forceinline__ unsigned enc_f32(float f) {
    unsigned u = __float_as_uint(f);
    return (u >> 31) ? ~u : (u | 0x80000000u);
}
__device__ __forceinline__ float dec_f32(unsigned u) {
    return (u >> 31) ? __uint_as_float(u & 0x7FFFFFFFu) : __uint_as_float(~u);
}

__global__ void fill_f32(float* p, float v, int count) {
    int i = blockIdx.x * blockDim.x + threadIdx.x;
    if (i < count) p[i] = v;
}
__global__ void fill_u32(unsigned* p, unsigned v, int count) {
    int i = blockIdx.x * blockDim.x + threadIdx.x;
    if (i < count) p[i] = v;
}

// ---------- WMMA f32 GEMM: OUT[M x N] = act(A[M x K] @ W[K x N] + bias) ----------
// One wave computes a 16x32 strip (two 16x16 WMMA tiles sharing the A fragment).
// Padded rows are CLAMPED, not zeroed: a padded A-row only feeds the same padded
// output row, which the store guard masks — so no predicated loads needed.
// row_map (optional) remaps logical row -> A row (fused gather).
__global__ __launch_bounds__(32)
void gemm_wmma_f32(const float* __restrict__ A, const float* __restrict__ W,
                   const float* __restrict__ bias, float* __restrict__ OUT,
                   const int* __restrict__ row_map, int M, int K, int N, int do_relu) {
    const int tile_m = blockIdx.x * 16;
    const int tile_n = blockIdx.y * 32;
    const int lane = threadIdx.x;        // 0..31
    const int half = lane >> 4;          // 0 or 1
    const int l15  = lane & 15;

    const int rowA = tile_m + l15;
    int rclamp = rowA < M ? rowA : M - 1;
    int rA = row_map ? row_map[rclamp] : rclamp;

    // per-lane base pointers; loop bumps by constants -> immediate-offset loads
    const float* ap  = A + (size_t)rA * K + half * 2;           // a.x/a.y contiguous
    const float* bp0 = W + (size_t)(half * 2) * N + tile_n + l15;
    const float* bp1 = bp0 + 16;
    const size_t bstep = (size_t)4 * N;

    v8f acc0 = {0.f, 0.f, 0.f, 0.f, 0.f, 0.f, 0.f, 0.f};
    v8f acc1 = {0.f, 0.f, 0.f, 0.f, 0.f, 0.f, 0.f, 0.f};

#pragma unroll 4
    for (int k0 = 0; k0 < K; k0 += 4) {
        v2f a, b0, b1;
        a.x  = ap[0];   a.y  = ap[1];
        b0.x = bp0[0];  b0.y = bp0[N];
        b1.x = bp1[0];  b1.y = bp1[N];
        acc0 = __builtin_amdgcn_wmma_f32_16x16x4_f32(false, a, false, b0,
                                                     (short)0, acc0, false, false);
        acc1 = __builtin_amdgcn_wmma_f32_16x16x4_f32(false, a, false, b1,
                                                     (short)0, acc1, false, false);
        ap += 4;
        bp0 += bstep;
        bp1 += bstep;
    }

    const int coln0 = tile_n + l15;
    const int coln1 = coln0 + 16;
    float bias0 = bias ? bias[coln0] : 0.f;
    float bias1 = bias ? bias[coln1] : 0.f;

#pragma unroll
    for (int r = 0; r < 8; ++r) {
        int row = tile_m + r + half * 8;
        if (row < M) {
            float v0 = acc0[r] + bias0;
            float v1 = acc1[r] + bias1;
            if (do_relu) {
                v0 = v0 > 0.f ? v0 : 0.f;
                v1 = v1 > 0.f ? v1 : 0.f;
            }
            OUT[(size_t)row * N + coln0] = v0;
            OUT[(size_t)row * N + coln1] = v1;
        }
    }
}

// ---------- per-node attention dots: es = h.a_s, ed = h.a_d ----------
__global__ __launch_bounds__(128)
void node_dots(const float* __restrict__ h, const float* __restrict__ a_s,
               const float* __restrict__ a_d, float* __restrict__ es,
               float* __restrict__ ed) {
    const int row = blockIdx.x;
    const int tid = threadIdx.x;  // 128
    float hv = h[(size_t)row * 128 + tid];
    __shared__ float sm1[128], sm2[128];
    sm1[tid] = hv * a_s[tid];
    sm2[tid] = hv * a_d[tid];
    __syncthreads();
    for (int off = 64; off > 0; off >>= 1) {
        if (tid < off) { sm1[tid] += sm1[tid + off]; sm2[tid] += sm2[tid + off]; }
        __syncthreads();
    }
    if (tid == 0) { es[row] = sm1[0]; ed[row] = sm2[0]; }
}

// ---------- edge pass A: logits + segment max (float atomicMax via uint encoding) ----------
__global__ void edge_logits(const int* __restrict__ ei, int E, int ntot,
                            const float* __restrict__ es, const float* __restrict__ ed,
                            float* __restrict__ val, unsigned* __restrict__ menc) {
    int e = blockIdx.x * blockDim.x + threadIdx.x;
    int ne = E + ntot;
    if (e >= ne) return;
    int s, d;
    if (e < E) { s = ei[e]; d = ei[E + e]; } else { s = d = e - E; }
    float v = es[s] + ed[d];
    v = v > 0.f ? v : 0.2f * v;          // leaky_relu(0.2)
    val[e] = v;
    atomicMax(&menc[d], enc_f32(v));
}

// ---------- edge pass B: exp(e - max) + segment sum ----------
__global__ void edge_weights(const int* __restrict__ ei, int E, int ntot,
                             float* __restrict__ val, const unsigned* __restrict__ menc,
                             float* __restrict__ z) {
    int e = blockIdx.x * blockDim.x + threadIdx.x;
    int ne = E + ntot;
    if (e >= ne) return;
    int d = (e < E) ? ei[E + e] : (e - E);
    float w = expf(val[e] - dec_f32(menc[d]));
    val[e] = w;
    atomicAdd(&z[d], w);
}

// ---------- edge pass C: out[dst] += h[src] * w ----------
__global__ __launch_bounds__(128)
void edge_scatter(const int* __restrict__ ei, int E, int ntot,
                  const float* __restrict__ h, const float* __restrict__ val,
                  float* __restrict__ out) {
    int e = blockIdx.x;
    int c = threadIdx.x;  // 128
    int s, d;
    if (e < E) { s = ei[e]; d = ei[E + e]; } else { s = d = e - E; }
    atomicAdd(&out[(size_t)d * 128 + c], h[(size_t)s * 128 + c] * val[e]);
}

// ---------- finalize: divide by z, add bias, relu, accumulate xsum ----------
__global__ void gat_finalize(float* __restrict__ xo, const float* __restrict__ z,
                             const float* __restrict__ b, float* __restrict__ xsum,
                             int n) {
    int i = blockIdx.x * blockDim.x + threadIdx.x;
    if (i >= n * 128) return;
    int row = i >> 7, c = i & 127;
    float v = xo[i] / (z[row] + 1e-16f) + b[c];
    v = v > 0.f ? v : 0.f;
    xo[i] = v;
    xsum[i] += v;
}

// ---------- i3_all: per-segment sum over target rows of h60 ----------
__global__ __launch_bounds__(256)
void seg_out(const float* __restrict__ h60, const int* __restrict__ seg,
             int ntgt, float* __restrict__ out) {
    int s = blockIdx.x, c = threadIdx.x;
    float acc = 0.f;
    for (int t = 0; t < ntgt; ++t)
        if (seg[t] == s) acc += h60[(size_t)t * 256 + c];
    out[113 + s * 256 + c] = acc;
}

// ---------- MLP head ----------
__global__ __launch_bounds__(256)
void build_pr(const float* __restrict__ emb0, const float* __restrict__ emb2,
              const int* __restrict__ adm0, int n0,
              const int* __restrict__ adm1, int n1, float* __restrict__ pr) {
    int c = threadIdx.x, s = blockIdx.x;  // 3 x 256
    float acc = 0.f;
    if (s == 0)       { for (int t = 0; t < n0; ++t) acc += emb0[(size_t)adm0[t] * 256 + c]; }
    else if (s == 1)  { for (int t = 0; t < n1; ++t) acc += emb0[(size_t)adm1[t] * 256 + c]; }
    else              { acc = emb2[112 * 256 + c]; }
    pr[s * 256 + c] = acc;
}

__global__ __launch_bounds__(256)
void mlp1(const float* __restrict__ pr, const float* __restrict__ Wr1,
          const float* __restrict__ br1, float* __restrict__ hh) {
    int c = blockIdx.x * 256 + threadIdx.x;  // 512 outputs
    float acc = br1[c];
    for (int k = 0; k < 768; ++k) acc += pr[k] * Wr1[(size_t)k * 512 + c];
    hh[c] = acc > 0.f ? acc : 0.f;
}

__global__ __launch_bounds__(128)
void mlp2(const float* __restrict__ hh, const float* __restrict__ Wr2,
          const float* __restrict__ br2, float* __restrict__ out) {
    int c = threadIdx.x;
    if (c >= 112) return;
    float acc = br2[c];
    for (int k = 0; k < 512; ++k) acc += hh[k] * Wr2[(size_t)k * 112 + c];
    out[c] = acc;
}

__global__ __launch_bounds__(128)
void ddi_kernel(const float* __restrict__ out, const float* __restrict__ ddi,
                float* __restrict__ dst) {
    int tid = threadIdx.x;  // 128
    __shared__ float p[112];
    __shared__ float red[128];
    if (tid < 112) p[tid] = 1.f / (1.f + expf(-out[tid]));
    __syncthreads();
    float acc = 0.f;
    for (int idx = tid; idx < 112 * 112; idx += 128) {
        int i = idx / 112, j = idx % 112;
        acc += p[i] * p[j] * ddi[idx];
    }
    red[tid] = acc;
    __syncthreads();
    for (int off = 64; off > 0; off >>= 1) {
        if (tid < off) red[tid] += red[tid + off];
        __syncthreads();
    }
    if (tid == 0) dst[0] = 0.0005f * red[0];
}

// ---------- launch ----------
extern "C" void kernel_launch(void* const* d_in, const int* in_sizes, int n_in,
                              void* d_out, int out_size, void* d_ws, size_t ws_size,
                              hipStream_t stream) {
    const float* emb0   = (const float*)d_in[0];
    const float* emb2   = (const float*)d_in[1];
    const float* kg_emb = (const float*)d_in[2];
    const float* W1  = (const float*)d_in[3];
    const float* as1 = (const float*)d_in[4];
    const float* ad1 = (const float*)d_in[5];
    const float* b1  = (const float*)d_in[6];
    const float* W2  = (const float*)d_in[7];
    const float* as2 = (const float*)d_in[8];
    const float* ad2 = (const float*)d_in[9];
    const float* b2  = (const float*)d_in[10];
    const float* W3  = (const float*)d_in[11];
    const float* as3 = (const float*)d_in[12];
    const float* ad3 = (const float*)d_in[13];
    const float* b3  = (const float*)d_in[14];
    const float* lin1_W = (const float*)d_in[15];
    const float* lin1_b = (const float*)d_in[16];
    const float* Wr1 = (const float*)d_in[17];
    const float* br1 = (const float*)d_in[18];
    const float* Wr2 = (const float*)d_in[19];
    const float* br2 = (const float*)d_in[20];
    const float* ddi = (const float*)d_in[21];
    const int* adm0  = (const int*)d_in[22]; int n0 = in_sizes[22];
    const int* adm1  = (const int*)d_in[23]; int n1 = in_sizes[23];
    const int* x_ids = (const int*)d_in[24]; int n  = in_sizes[24];
    const int* ei    = (const int*)d_in[25]; int E  = in_sizes[25] / 2;
    const int* tidx  = (const int*)d_in[26]; int ntgt = in_sizes[26];
    const int* tseg  = (const int*)d_in[27];
    float* out = (float*)d_out;
    int nseg = (out_size - 113) / 256;   // i3_all rows (== num_subgraphs)
    int ne = E + n;                      // edges incl. self loops

    // --- carve workspace ---
    size_t off = 0;
    char* base = (char*)d_ws;
    auto carve = [&](size_t bytes) -> void* {
        void* p = base + off;
        off += (bytes + 255) & ~(size_t)255;
        return p;
    };
    float* feat1 = (float*)carve((size_t)n * 128 * 4);
    float* feat2 = (float*)carve((size_t)n * 128 * 4);
    float* hbuf  = (float*)carve((size_t)n * 128 * 4);
    float* xsum  = (float*)carve((size_t)n * 128 * 4);
    float* es    = (float*)carve((size_t)n * 4);
    float* ed    = (float*)carve((size_t)n * 4);
    float* zbuf  = (float*)carve((size_t)n * 4);
    unsigned* menc = (unsigned*)carve((size_t)n * 4);
    float* val   = (float*)carve((size_t)ne * 4);
    float* h60   = (float*)carve((size_t)((ntgt + 15) & ~15) * 256 * 4);
    float* pr    = (float*)carve(768 * 4);
    float* hh    = (float*)carve(512 * 4);
    (void)ws_size; (void)n_in;

    // --- MLP head (independent of GNN) ---
    build_pr<<<3, 256, 0, stream>>>(emb0, emb2, adm0, n0, adm1, n1, pr);
    mlp1<<<2, 256, 0, stream>>>(pr, Wr1, br1, hh);
    mlp2<<<1, 128, 0, stream>>>(hh, Wr2, br2, out);
    ddi_kernel<<<1, 128, 0, stream>>>(out, ddi, out + 112);

    // --- GNN ---
    int nf = n * 128;
    fill_f32<<<(nf + 255) / 256, 256, 0, stream>>>(xsum, 0.f, nf);

    const float* Ain[3]  = {kg_emb, feat1, feat2};
    const int*   rmap[3] = {x_ids, nullptr, nullptr};
    float*       Aout[3] = {feat1, feat2, feat1};
    const float* Ws[3]   = {W1, W2, W3};
    const float* As[3]   = {as1, as2, as3};
    const float* Ad[3]   = {ad1, ad2, ad3};
    const float* Bs[3]   = {b1, b2, b3};
    int Kdim[3] = {256, 128, 128};

    for (int l = 0; l < 3; ++l) {
        float* xo = Aout[l];
        fill_f32<<<(nf + 255) / 256, 256, 0, stream>>>(xo, 0.f, nf);
        fill_f32<<<(n + 255) / 256, 256, 0, stream>>>(zbuf, 0.f, n);
        fill_u32<<<(n + 255) / 256, 256, 0, stream>>>(menc, 0u, n);

        dim3 g((n + 15) / 16, 128 / 32);
        gemm_wmma_f32<<<g, 32, 0, stream>>>(Ain[l], Ws[l], nullptr, hbuf,
                                            rmap[l], n, Kdim[l], 128, 0);
        node_dots<<<n, 128, 0, stream>>>(hbuf, As[l], Ad[l], es, ed);
        edge_logits<<<(ne + 255) / 256, 256, 0, stream>>>(ei, E, n, es, ed, val, menc);
        edge_weights<<<(ne + 255) / 256, 256, 0, stream>>>(ei, E, n, val, menc, zbuf);
        edge_scatter<<<ne, 128, 0, stream>>>(ei, E, n, hbuf, val, xo);
        gat_finalize<<<(nf + 255) / 256, 256, 0, stream>>>(xo, zbuf, Bs[l], xsum, n);
    }

    // lin1 on target rows only (fused gather via row_map), then per-segment sum
    dim3 g60((ntgt + 15) / 16, 256 / 32);
    gemm_wmma_f32<<<g60, 32, 0, stream>>>(xsum, lin1_W, lin1_b, h60,
                                          tidx, ntgt, 128, 256, 1);
    seg_out<<<nseg, 256, 0, stream>>>(h60, tseg, ntgt, out);
}